// KGEScorer_66460323938768
// MI455X (gfx1250) — compile-verified
//
#include <hip/hip_runtime.h>
#include <hip/hip_bf16.h>

// ---------------- problem constants (from reference) ----------------
constexpr int B = 4;
constexpr int TG = 32768;
constexpr int M = 4;
constexpr int D = 128;
constexpr int K_OUT = 2048;           // OUTPUT_BUDGET
constexpr float BIGF = 1e9f;
constexpr int G_TOTAL = B * TG;        // 131072 groundings
constexpr int ATOMS = G_TOTAL * M;     // 524288 atoms
constexpr int TILE_ATOMS = 16;         // atoms per wave (4 groundings)
constexpr int TILES = ATOMS / TILE_ATOMS;   // 32768
constexpr int WAVES_PER_BLOCK = 2;
constexpr int LDS_ROW = D + 4;         // 132 floats: bank-conflict-free (4*row+k)

typedef float v2f __attribute__((ext_vector_type(2)));
typedef float v8f __attribute__((ext_vector_type(8)));

__device__ __forceinline__ float swz_f32(float x, int imm) {
    // ds_swizzle group-of-32: offset = xor<<10 | or<<5 | and
    int r;
    if (imm == 0x41F) r = __builtin_amdgcn_ds_swizzle(__float_as_int(x), 0x41F); // XOR 1
    else              r = __builtin_amdgcn_ds_swizzle(__float_as_int(x), 0x81F); // XOR 2
    return __int_as_float(r);
}

// ---------------------------------------------------------------
// Kernel 1: score all groundings.
// One wave = 16 atoms = 4 groundings. Stage p=s*r and o into LDS with
// coalesced b128 row loads, then 32 chained V_WMMA_F32_16X16X4_F32 compute
// the 16 dot products as the diagonal of (P x O^T).
// ---------------------------------------------------------------
__global__ void __launch_bounds__(WAVES_PER_BLOCK * 32)
kge_score_kernel(const int* __restrict__ body,      // [B*TG*M*3]
                 const int* __restrict__ mask,      // [B*TG]
                 const float* __restrict__ ent_emb, // [N_ENT*D]
                 const float* __restrict__ rel_emb, // [N_REL*D]
                 float* __restrict__ scores)        // [B*TG]
{
    __shared__ __align__(16) float pls[WAVES_PER_BLOCK][16 * LDS_ROW];
    __shared__ __align__(16) float ols[WAVES_PER_BLOCK][16 * LDS_ROW];

    const int lane = threadIdx.x & 31;
    const int wave = threadIdx.x >> 5;
    const int tile = blockIdx.x * WAVES_PER_BLOCK + wave;   // [0, TILES)
    const int g0   = tile * 4;                              // first grounding
    const int a0   = tile * TILE_ATOMS;                     // first atom

    float* __restrict__ pbase = pls[wave];
    float* __restrict__ obase = ols[wave];

    // ---- per-row atom ids held by lanes 0..15 ----
    int row_subj = 0, row_pred = 0, row_obj = 0;
    if (lane < 16) {
        const int* bp = body + (size_t)(a0 + lane) * 3;
        row_subj = bp[0];
        row_pred = bp[1];
        row_obj  = bp[2];
    }
    // bit m set => atom m is active (subject != padding)
    const unsigned act_mask = __builtin_amdgcn_ballot_w32((lane < 16) && (row_subj != 0));

    // ---- stage p = s*r and o into LDS (coalesced b128 per row) ----
    #pragma unroll 4
    for (int row = 0; row < 16; ++row) {
        const int subj = __builtin_amdgcn_readlane(row_subj, row);
        const int pred = __builtin_amdgcn_readlane(row_pred, row);
        const int obj  = __builtin_amdgcn_readlane(row_obj,  row);
        const float4 sv = *(const float4*)(ent_emb + (size_t)subj * D + lane * 4);
        const float4 rv = *(const float4*)(rel_emb + (size_t)pred * D + lane * 4);
        const float4 ov = *(const float4*)(ent_emb + (size_t)obj  * D + lane * 4);
        float4 pv;
        pv.x = sv.x * rv.x; pv.y = sv.y * rv.y;
        pv.z = sv.z * rv.z; pv.w = sv.w * rv.w;
        *(float4*)(pbase + row * LDS_ROW + lane * 4) = pv;
        *(float4*)(obase + row * LDS_ROW + lane * 4) = ov;
    }
    __syncthreads();

    // ---- WMMA: C += A(16x4) x B(4x16), K looped over 128 ----
    // f32 16x4 A layout: lanes 0-15 -> row=lane, K=kb+{0,1}; lanes 16-31 -> row=lane-16, K=kb+{2,3}
    // f32 4x16 B layout: lanes 0-15 -> col=lane, K=kb+{0,1}; lanes 16-31 -> col=lane-16, K=kb+{2,3}
    const int rc   = lane & 15;                 // row (A) / col (B)
    const int koff = (lane < 16) ? 0 : 2;
    const float* __restrict__ prow = pbase + rc * LDS_ROW + koff;
    const float* __restrict__ orow = obase + rc * LDS_ROW + koff;

    v8f c = {0.f, 0.f, 0.f, 0.f, 0.f, 0.f, 0.f, 0.f};
    #pragma unroll
    for (int kb = 0; kb < D; kb += 4) {
        const v2f a = *(const v2f*)(prow + kb);   // 8B aligned
        const v2f b = *(const v2f*)(orow + kb);
        c = __builtin_amdgcn_wmma_f32_16x16x4_f32(
                /*neg_a=*/false, a, /*neg_b=*/false, b,
                /*c_mod=*/(short)0, c, /*reuse_a=*/false, /*reuse_b=*/false);
    }

    // ---- extract diagonal: atom m lives at lane m (m<8) / lane m+16 (m>=8) ----
    // Diagonal-bearing lanes: 0..7 (element c[lane]) and 24..31 (element c[lane-24]).
    const bool hasdiag = (lane < 8) || (lane >= 24);
    const int  ii = lane & 7;
    float dval = c[0];
    dval = (ii == 1) ? c[1] : dval;
    dval = (ii == 2) ? c[2] : dval;
    dval = (ii == 3) ? c[3] : dval;
    dval = (ii == 4) ? c[4] : dval;
    dval = (ii == 5) ? c[5] : dval;
    dval = (ii == 6) ? c[6] : dval;
    dval = (ii == 7) ? c[7] : dval;

    const int m_atom = (lane < 8) ? lane : (lane - 16);     // valid on diag lanes
    float atom_score = ((act_mask >> (m_atom & 15)) & 1u) ? dval : BIGF;
    if (!hasdiag) atom_score = BIGF;   // keep non-diag lanes inert

    // ---- min over 4 atoms: diag lanes sit in groups {0-3},{4-7},{24-27},{28-31} ----
    atom_score = fminf(atom_score, swz_f32(atom_score, 0x41F)); // XOR 1
    atom_score = fminf(atom_score, swz_f32(atom_score, 0x81F)); // XOR 2

    if (hasdiag && ((lane & 3) == 0)) {
        const int g_local = (lane < 8) ? (lane >> 2) : ((lane - 16) >> 2);
        const int G = g0 + g_local;
        const float s = mask[G] ? atom_score : -BIGF;
        scores[G] = s;
    }
}

// ---------------------------------------------------------------
// Kernel 2: exact sorted top-K per batch row.
// 32768 keys (~mono(score)<<32 | idx) sorted ascending by in-LDS bitonic sort
// (256 KB dynamic LDS, fits CDNA5's 320 KB/WG). Ascending key order ==
// descending score with ascending index tie-break == jax.lax.top_k order.
// ---------------------------------------------------------------
__global__ void __launch_bounds__(1024)
kge_topk_kernel(const float* __restrict__ scores,   // [B*TG]
                int* __restrict__ topidx)           // [B*K_OUT]
{
    extern __shared__ unsigned long long keys[];    // TG entries
    const int row = blockIdx.x;
    const int tid = threadIdx.x;
    const int T   = blockDim.x;

    for (int i = tid; i < TG; i += T) {
        const unsigned u = __float_as_uint(scores[row * TG + i]);
        const unsigned mono = u ^ ((u >> 31) ? 0xFFFFFFFFu : 0x80000000u);
        keys[i] = ((unsigned long long)(~mono) << 32) | (unsigned)i;
    }
    __syncthreads();

    for (unsigned k = 2; k <= (unsigned)TG; k <<= 1) {
        for (unsigned j = k >> 1; j > 0; j >>= 1) {
            for (unsigned i = tid; i < (unsigned)TG; i += T) {
                const unsigned ixj = i ^ j;
                if (ixj > i) {
                    const bool up = ((i & k) == 0);
                    const unsigned long long a = keys[i];
                    const unsigned long long b = keys[ixj];
                    if ((a > b) == up) { keys[i] = b; keys[ixj] = a; }
                }
            }
            __syncthreads();
        }
    }

    for (int i = tid; i < K_OUT; i += T)
        topidx[row * K_OUT + i] = (int)(keys[i] & 0xFFFFFFFFu);
}

// ---------------------------------------------------------------
// Kernel 3: gather selected rows into concatenated output (float buffer).
// Layout: body_sel [B,K,M,3] | mask_sel [B,K] | rule_sel [B,K] | top_scores [B,K]
// ---------------------------------------------------------------
__global__ void __launch_bounds__(256)
kge_gather_kernel(const int* __restrict__ body,
                  const int* __restrict__ mask,
                  const int* __restrict__ rule_idx,
                  const float* __restrict__ scores,
                  const int* __restrict__ topidx,
                  float* __restrict__ out)
{
    const int t = blockIdx.x * blockDim.x + threadIdx.x;
    if (t >= B * K_OUT) return;
    const int b = t / K_OUT;
    const int k = t % K_OUT;
    const int tg = topidx[b * K_OUT + k];

    constexpr int BODY_N   = B * K_OUT * M * 3;     // 98304
    constexpr int MASK_OFF = BODY_N;                // 98304
    constexpr int RULE_OFF = MASK_OFF + B * K_OUT;  // 106496
    constexpr int SCR_OFF  = RULE_OFF + B * K_OUT;  // 114688

    const int gsrc = b * TG + tg;
    #pragma unroll
    for (int m = 0; m < M; ++m) {
        #pragma unroll
        for (int j = 0; j < 3; ++j) {
            out[((b * K_OUT + k) * M + m) * 3 + j] =
                (float)body[((size_t)gsrc * M + m) * 3 + j];
        }
    }
    out[MASK_OFF + t] = mask[gsrc] ? 1.0f : 0.0f;
    out[RULE_OFF + t] = (float)rule_idx[gsrc];
    out[SCR_OFF  + t] = scores[gsrc];
}

// ---------------------------------------------------------------
extern "C" void kernel_launch(void* const* d_in, const int* in_sizes, int n_in,
                              void* d_out, int out_size, void* d_ws, size_t ws_size,
                              hipStream_t stream) {
    const int*   body     = (const int*)d_in[0];   // [B,TG,M,3]
    const int*   mask     = (const int*)d_in[1];   // [B,TG]
    const int*   rule_idx = (const int*)d_in[2];   // [B,TG]
    const float* ent_emb  = (const float*)d_in[3]; // [N_ENT,D]
    const float* rel_emb  = (const float*)d_in[4]; // [N_REL,D]
    float* out = (float*)d_out;

    float* scores = (float*)d_ws;                                  // B*TG floats
    int*   topidx = (int*)((char*)d_ws + (size_t)B * TG * sizeof(float)); // B*K_OUT ints

    const int score_blocks = TILES / WAVES_PER_BLOCK;              // 16384
    kge_score_kernel<<<score_blocks, WAVES_PER_BLOCK * 32, 0, stream>>>(
        body, mask, ent_emb, rel_emb, scores);

    kge_topk_kernel<<<B, 1024, TG * sizeof(unsigned long long), stream>>>(
        scores, topidx);

    const int gth = B * K_OUT;
    kge_gather_kernel<<<(gth + 255) / 256, 256, 0, stream>>>(
        body, mask, rule_idx, scores, topidx, out);
}